// Normalized_Correlation_Layer_81561428951588
// MI455X (gfx1250) — compile-verified
//
#include <hip/hip_runtime.h>
#include <math.h>

// Problem constants (from reference)
#define BATCH 4
#define HDIM 20
#define WDIM 20
#define CDIM 25
#define OR 16
#define OC 16
#define NPAT 6400      // CDIM * OR * OC
#define PLEN 25        // KH*KW
#define KPAD 28        // PLEN padded to multiple of 4 for wmma 16x16x4
#define MTILES 400     // NPAT / 16
#define NCHUNKS 16     // n-tiles split into 16 strips of 25 tiles
#define TILES_PER_STRIP 25

typedef float v2f __attribute__((ext_vector_type(2)));
typedef float v8f __attribute__((ext_vector_type(8)));

// ---------------------------------------------------------------------------
// Phase 1: per-patch normalization, written TRANSPOSED: xt[b][k][n], k in [0,28)
// One thread per (b, n) patch.  B*N = 25600 threads total.
// ---------------------------------------------------------------------------
__global__ __launch_bounds__(256) void normalize_patches_k(
    const float* __restrict__ x, float* __restrict__ xt) {
  const int t = blockIdx.x * 256 + threadIdx.x;     // 0 .. 25599 (grid exact)
  const int b = t / NPAT;
  const int n = t % NPAT;
  const int k   = n >> 8;      // channel  (n / 256)
  const int rem = n & 255;
  const int i = rem >> 4;      // patch row
  const int j = rem & 15;      // patch col

  float v[PLEN];
  float sum = 0.f, sq = 0.f;
#pragma unroll
  for (int di = 0; di < 5; ++di) {
#pragma unroll
    for (int dj = 0; dj < 5; ++dj) {
      const float val = x[((b * HDIM + i + di) * WDIM + (j + dj)) * CDIM + k];
      v[di * 5 + dj] = val;
      sum += val;
      sq  += val * val;
    }
  }
  const float m   = sum * (1.0f / 25.0f);
  const float var = sq * (1.0f / 25.0f) - m * m;
  const float inv = 1.0f / sqrtf(var);   // population std, matches K.std

  const int base = b * (KPAD * NPAT) + n;
#pragma unroll
  for (int p = 0; p < PLEN; ++p) xt[base + p * NPAT] = (v[p] - m) * inv;
#pragma unroll
  for (int p = PLEN; p < KPAD; ++p) xt[base + p * NPAT] = 0.0f;
}

// ---------------------------------------------------------------------------
// Phase 2: cost volume via V_WMMA_F32_16X16X4_F32.
// D[b] = X2[b] (6400x25) * X1[b]^T (25x6400), K padded to 28 (7 wmma/tile).
//
// One wave per (b, m-tile, n-strip).  Waves total = 4*400*16 = 25600
//   -> grid 3200 blocks x 256 threads (8 waves/block), exact, no divergence.
//
// A fragment (16x4 f32): lanes 0-15 hold M=0..15 with K=0 (v0), K=1 (v1);
//                        lanes 16-31 hold K=2 (v0), K=3 (v1).
// B fragment mirrors A with N as the per-lane row.
// C/D (16x16 f32, 8 VGPRs): VGPR r -> M = r + 8*(lane>=16), N = lane&15.
// ---------------------------------------------------------------------------
__global__ __launch_bounds__(256) void corr_wmma_k(
    const float* __restrict__ x2t,   // A source (rows n2), transposed [b][k][n]
    const float* __restrict__ x1t,   // B source (cols n1), transposed [b][k][n]
    float* __restrict__ out) {
  const int lane   = threadIdx.x & 31;
  const int wid    = (blockIdx.x * 256 + threadIdx.x) >> 5;
  const int b      = wid / (MTILES * NCHUNKS);
  const int r      = wid % (MTILES * NCHUNKS);
  const int mt     = r / NCHUNKS;
  const int nchunk = r % NCHUNKS;

  const int m0    = mt << 4;
  const int lrow  = lane & 15;          // per-lane row (M for A, N for B)
  const int khalf = (lane >> 4) << 1;   // lanes 0-15 -> K 0/1, 16-31 -> K 2/3
  const int hi    = (lane >> 4) << 3;   // C/D: upper half lanes hold M+8

  const float* abase = x2t + b * (KPAD * NPAT);
  const float* bbase = x1t + b * (KPAD * NPAT);

  // Load the 7 A fragments once; reused across the whole 25-tile strip.
  v2f a[7];
#pragma unroll
  for (int s = 0; s < 7; ++s) {
    const float* p = abase + (4 * s + khalf) * NPAT + m0 + lrow;
    a[s].x = p[0];
    a[s].y = p[NPAT];
  }

  float* outb = out + (size_t)b * NPAT * NPAT;

  for (int nt = 0; nt < TILES_PER_STRIP; ++nt) {
    const int n0 = (nchunk * TILES_PER_STRIP + nt) << 4;

    v8f c = {0.f, 0.f, 0.f, 0.f, 0.f, 0.f, 0.f, 0.f};
#pragma unroll
    for (int s = 0; s < 7; ++s) {
      const float* p = bbase + (4 * s + khalf) * NPAT + n0 + lrow;
      v2f bf;
      bf.x = p[0];
      bf.y = p[NPAT];
      // (neg_a, A, neg_b, B, c_mod, C, reuse_a, reuse_b)
      c = __builtin_amdgcn_wmma_f32_16x16x4_f32(
          false, a[s], false, bf, (short)0, c, false, false);
    }

    float* op = outb + (size_t)(m0 + hi) * NPAT + n0 + lrow;
#pragma unroll
    for (int rr = 0; rr < 8; ++rr) op[(size_t)rr * NPAT] = c[rr];
  }
}

// ---------------------------------------------------------------------------
extern "C" void kernel_launch(void* const* d_in, const int* in_sizes, int n_in,
                              void* d_out, int out_size, void* d_ws, size_t ws_size,
                              hipStream_t stream) {
  const float* in1 = (const float*)d_in[0];
  const float* in2 = (const float*)d_in[1];
  float* xt1 = (float*)d_ws;                       // [4][28][6400]
  float* xt2 = xt1 + BATCH * KPAD * NPAT;          // [4][28][6400]

  // Phase 1: 25600 threads = 100 blocks x 256 (exact)
  normalize_patches_k<<<(BATCH * NPAT) / 256, 256, 0, stream>>>(in1, xt1);
  normalize_patches_k<<<(BATCH * NPAT) / 256, 256, 0, stream>>>(in2, xt2);

  // Phase 2: 25600 waves = 3200 blocks x 256 (exact)
  corr_wmma_k<<<(BATCH * MTILES * NCHUNKS) / 8, 256, 0, stream>>>(
      xt2, xt1, (float*)d_out);
}